// TopDownHTMM_39762807227044
// MI455X (gfx1250) — compile-verified
//
#include <hip/hip_runtime.h>
#include <math.h>

typedef __attribute__((ext_vector_type(2))) float v2f;
typedef __attribute__((ext_vector_type(8))) float v8f;

#define T_   128
#define D_   10
#define C_   8
#define G_   16
#define M_   256
#define N1_  2047          // nodes per tree
#define DIM_ 262016        // T_ * N1_

static __device__ __forceinline__ v2f ld2(const float* p) { return *(const v2f*)p; }

// D(16x16,f32) = A(16x4,f32) x B(4x16,f32) + C   -- v_wmma_f32_16x16x4_f32
static __device__ __forceinline__ v8f wmma4(v2f a, v2f b, v8f c) {
  return __builtin_amdgcn_wmma_f32_16x16x4_f32(
      /*neg_a=*/false, a, /*neg_b=*/false, b,
      /*c_mod=*/(short)0, c, /*reuse_a=*/false, /*reuse_b=*/false);
}

// ---------- softmax of A (axis 0 over child state j) and Pi (axis 0) ----------
__global__ void k_softmax_small(const float* __restrict__ A, const float* __restrict__ Pi,
                                float* __restrict__ smA, float* __restrict__ smAT,
                                float* __restrict__ smPi) {
  const int tid = threadIdx.x;
  if (tid < C_ * G_) {               // one thread per (i,g) column of A
    const int i = tid >> 4, g = tid & 15;
    float v[C_], mx = -1e30f;
    for (int j = 0; j < C_; ++j) { v[j] = A[(j * C_ + i) * G_ + g]; mx = fmaxf(mx, v[j]); }
    float s = 0.f;
    for (int j = 0; j < C_; ++j) { v[j] = expf(v[j] - mx); s += v[j]; }
    const float inv = 1.0f / s;
    for (int j = 0; j < C_; ++j) {
      const float e = v[j] * inv;
      smA [g * 64 + j * 8 + i] = e;   // [g][j][i]
      smAT[g * 64 + i * 8 + j] = e;   // [g][i][j] = transpose
    }
  } else if (tid < C_ * G_ + G_) {   // one thread per g for Pi
    const int g = tid - C_ * G_;
    float v[C_], mx = -1e30f;
    for (int c = 0; c < C_; ++c) { v[c] = Pi[c * G_ + g]; mx = fmaxf(mx, v[c]); }
    float s = 0.f;
    for (int c = 0; c < C_; ++c) { v[c] = expf(v[c] - mx); s += v[c]; }
    const float inv = 1.0f / s;
    for (int c = 0; c < C_; ++c) smPi[g * 8 + c] = v[c] * inv;
  }
}

// ---------- softmax of B over symbol axis m; store as [g][m][c] ----------
__global__ void k_softmax_B(const float* __restrict__ B, float* __restrict__ smB) {
  const int tid = blockIdx.x * blockDim.x + threadIdx.x;
  if (tid >= C_ * G_) return;
  const int c = tid >> 4, g = tid & 15;
  float mx = -1e30f;
  for (int m = 0; m < M_; ++m) mx = fmaxf(mx, B[(c * M_ + m) * G_ + g]);
  float s = 0.f;
  for (int m = 0; m < M_; ++m) s += expf(B[(c * M_ + m) * G_ + g] - mx);
  const float inv = 1.0f / s;
  for (int m = 0; m < M_; ++m)
    smB[(size_t)g * (M_ * 8) + m * 8 + c] = expf(B[(c * M_ + m) * G_ + g] - mx) * inv;
}

// ---------- roots: prior[root] = smPi ----------
__global__ void k_init_roots(const float* __restrict__ smPi, float* __restrict__ prior) {
  const int tid = blockIdx.x * blockDim.x + threadIdx.x;
  if (tid >= T_ * G_ * C_) return;
  const int c = tid & 7;
  const int t = (tid >> 3) % T_;
  const int g = tid / (T_ * C_);
  prior[(size_t)g * DIM_ * 8 + (size_t)t * N1_ * 8 + c] = smPi[g * 8 + c];
}

// ---------- downward level d: prior[ch] = A_g * prior[pa] (both children identical) ----------
__global__ __launch_bounds__(32) void k_down(const float* __restrict__ smA,
                                             float* __restrict__ prior, int d) {
  const int g = blockIdx.y;
  const float* smA_g = smA + g * 64;
  float* pg = prior + (size_t)g * (size_t)DIM_ * 8;

  const int lane = threadIdx.x;
  const int t = lane & 15, hi = lane >> 4;
  const int q = blockIdx.x * 16 + t;            // parent index within level (all trees)
  const int tree = q >> d;
  const int local = ((1 << d) - 1) + (q & ((1 << d) - 1));
  const size_t npa = (size_t)tree * N1_ + local;

  // B operand: parent vector, K = i = {chunk*4 + 2*hi + 0,1}
  const float* in = pg + npa * 8;
  const v2f b0 = ld2(in + 2 * hi);
  const v2f b1 = ld2(in + 4 + 2 * hi);

  // A operand: rows j = lane&15 (rows >= 8 are zero padding)
  v2f a0 = {0.f, 0.f}, a1 = {0.f, 0.f};
  if (t < 8) { a0 = ld2(smA_g + t * 8 + 2 * hi); a1 = ld2(smA_g + t * 8 + 4 + 2 * hi); }

  v8f acc = {0.f, 0.f, 0.f, 0.f, 0.f, 0.f, 0.f, 0.f};
  acc = wmma4(a0, b0, acc);
  acc = wmma4(a1, b1, acc);

  if (lane < 16) {                              // lane L holds full 8-vector of parent q0+L
    const size_t nc = (size_t)tree * N1_ + 2 * (size_t)local + 1;  // children adjacent
    *(v8f*)(pg + nc * 8)     = acc;
    *(v8f*)(pg + nc * 8 + 8) = acc;
  }
}

// ---------- beta init: beta = prior * emission; leaves normalized + ll ----------
__global__ void k_init_beta(const float* __restrict__ prior, const float* __restrict__ smB,
                            const int* __restrict__ x, float* __restrict__ beta,
                            float* __restrict__ ll) {
  const size_t n = (size_t)blockIdx.x * blockDim.x + threadIdx.x;
  if (n >= (size_t)DIM_) return;
  const int g = blockIdx.y;
  const float* pg = prior + (size_t)g * DIM_ * 8;
  float* bg = beta + (size_t)g * DIM_ * 8;
  const float* smB_g = smB + (size_t)g * M_ * 8;

  const int m = x[n];
  const v8f p = *(const v8f*)(pg + n * 8);
  const v8f e = *(const v8f*)(smB_g + (size_t)m * 8);
  v8f b = p * e;
  const int local = (int)(n % (size_t)N1_);
  if (local >= (1 << D_) - 1) {                 // leaf: normalize, record log-evidence
    float nu = 0.f;
    for (int c = 0; c < 8; ++c) nu += b[c];
    b = b * (1.0f / nu);
    ll[(size_t)g * DIM_ + n] = logf(nu);
  }
  *(v8f*)(bg + n * 8) = b;
}

// ---------- upward level d ----------
__global__ __launch_bounds__(32) void k_up(const float* __restrict__ smAT,
                                           const float* __restrict__ prior,
                                           float* __restrict__ beta,
                                           float* __restrict__ ll, int d) {
  const int g = blockIdx.y;
  const float* at_g = smAT + g * 64;
  const float* pg = prior + (size_t)g * DIM_ * 8;
  float* bg = beta + (size_t)g * DIM_ * 8;
  float* llg = ll + (size_t)g * DIM_;

  const int lane = threadIdx.x;
  const int t = lane & 15, hi = lane >> 4;
  const int q = blockIdx.x * 16 + t;
  const int tree = q >> d;
  const int local = ((1 << d) - 1) + (q & ((1 << d) - 1));
  const size_t npa = (size_t)tree * N1_ + local;
  const size_t nc0 = (size_t)tree * N1_ + 2 * (size_t)local + 1;  // first child
  const size_t nc1 = nc0 + 1;                                      // second child

  // A operand: A_g^T rows i = lane&15 (rows >= 8 zero)
  v2f a0 = {0.f, 0.f}, a1 = {0.f, 0.f};
  if (t < 8) { a0 = ld2(at_g + t * 8 + 2 * hi); a1 = ld2(at_g + t * 8 + 4 + 2 * hi); }

  // B operands: r = beta/prior per child, components K = j
  const v2f r00 = ld2(bg + nc0 * 8 + 2 * hi)     / ld2(pg + nc0 * 8 + 2 * hi);
  const v2f r01 = ld2(bg + nc0 * 8 + 4 + 2 * hi) / ld2(pg + nc0 * 8 + 4 + 2 * hi);
  const v2f r10 = ld2(bg + nc1 * 8 + 2 * hi)     / ld2(pg + nc1 * 8 + 2 * hi);
  const v2f r11 = ld2(bg + nc1 * 8 + 4 + 2 * hi) / ld2(pg + nc1 * 8 + 4 + 2 * hi);

  v8f uv0 = {0.f, 0.f, 0.f, 0.f, 0.f, 0.f, 0.f, 0.f};
  uv0 = wmma4(a0, r00, uv0);
  uv0 = wmma4(a1, r01, uv0);
  v8f uv1 = {0.f, 0.f, 0.f, 0.f, 0.f, 0.f, 0.f, 0.f};
  uv1 = wmma4(a0, r10, uv1);
  uv1 = wmma4(a1, r11, uv1);
  const v8f prod = uv0 * uv1;                    // scatter-mul over the 2 children

  if (lane < 16) {                               // lane L: parent q0+L, states i=0..7
    const v8f bpa = *(const v8f*)(bg + npa * 8);
    v8f bu = bpa * prod;
    float nu = 0.f;
    for (int c = 0; c < 8; ++c) nu += bu[c];
    const v8f bn = bu * (1.0f / nu);
    *(v8f*)(bg + npa * 8) = bn;
    llg[npa] = logf(nu);
  }
}

// ---------- per-(tree,g) segment sum of ll ----------
__global__ void k_reduce(const float* __restrict__ ll, float* __restrict__ out) {
  const int t = blockIdx.x, g = blockIdx.y;
  const float* llg = ll + (size_t)g * DIM_ + (size_t)t * N1_;
  __shared__ float sh[128];
  float s = 0.f;
  for (int i = threadIdx.x; i < N1_; i += 128) s += llg[i];
  sh[threadIdx.x] = s;
  __syncthreads();
  for (int w = 64; w > 0; w >>= 1) {
    if (threadIdx.x < w) sh[threadIdx.x] += sh[threadIdx.x + w];
    __syncthreads();
  }
  if (threadIdx.x == 0) out[t * G_ + g] = sh[0];
}

extern "C" void kernel_launch(void* const* d_in, const int* in_sizes, int n_in,
                              void* d_out, int out_size, void* d_ws, size_t ws_size,
                              hipStream_t stream) {
  (void)in_sizes; (void)n_in; (void)out_size; (void)ws_size;
  const float* A  = (const float*)d_in[0];   // (C,C,G)
  const float* B  = (const float*)d_in[1];   // (C,M,G)
  const float* Pi = (const float*)d_in[2];   // (C,G)
  const int*   x  = (const int*)d_in[3];     // (DIM,)
  // d_in[4] = trees_ind: implicit in BFS layout, unused
  float* out = (float*)d_out;                // (T,G) f32

  float* ws    = (float*)d_ws;
  float* smA   = ws;                                   // G*64
  float* smAT  = smA  + 1024;                          // G*64
  float* smPi  = smAT + 1024;                          // G*8
  float* smB   = smPi + 128;                           // G*M*8
  float* prior = smB  + (size_t)G_ * M_ * 8;           // G*DIM*8
  float* beta  = prior + (size_t)G_ * DIM_ * 8;        // G*DIM*8
  float* ll    = beta  + (size_t)G_ * DIM_ * 8;        // G*DIM

  k_softmax_small<<<1, 160, 0, stream>>>(A, Pi, smA, smAT, smPi);
  k_softmax_B<<<1, 128, 0, stream>>>(B, smB);
  k_init_roots<<<(T_ * G_ * C_ + 255) / 256, 256, 0, stream>>>(smPi, prior);

  for (int d = 0; d < D_; ++d)
    k_down<<<dim3(8 << d, G_), 32, 0, stream>>>(smA, prior, d);

  k_init_beta<<<dim3((DIM_ + 255) / 256, G_), 256, 0, stream>>>(prior, smB, x, beta, ll);

  for (int d = D_ - 1; d >= 0; --d)
    k_up<<<dim3(8 << d, G_), 32, 0, stream>>>(smAT, prior, beta, ll, d);

  k_reduce<<<dim3(T_, G_), 128, 0, stream>>>(ll, out);
}